// TotalRegistrationLoss_22565758173292
// MI455X (gfx1250) — compile-verified
//
#include <hip/hip_runtime.h>
#include <math.h>

// TotalRegistrationLoss for MI455X (gfx1250, wave32).
// Latency-bound gather + reduction; WMMA f32 16x16x4 used as the
// cross-lane reduction engine (column-sum trick with all-ones A).

typedef __attribute__((ext_vector_type(2))) float v2f;
typedef __attribute__((ext_vector_type(8))) float v8f;

#define DIM_D 224
#define DIM_H 224
#define DIM_W 224
#define N_LM  2048
#define NTHREADS 1024   // 32 wave32 waves on one WGP

// Reduce one float per lane across the wave using V_WMMA_F32_16X16X4_F32.
// B (4x16 f32, 2 VGPRs): lane L's value goes into column L%16 of some row
// (VGPR0); VGPR1 rows are zero. With A = ones(16x4), D[m][n] = colsum(n)
// = s[n] + s[n+16], replicated across all rows -> every lane's d[0] holds
// colsum(lane%16). Four shfl_xor steps sum the 16 distinct columns.
// Requires EXEC == all ones (call only from uniform control flow).
__device__ __forceinline__ float wave_reduce_wmma(float s) {
  v2f a; a[0] = 1.0f; a[1] = 1.0f;          // all-ones A (layout-independent)
  v2f b; b[0] = s;    b[1] = 0.0f;          // one value per lane, rest zero
  v8f c = {};
  c = __builtin_amdgcn_wmma_f32_16x16x4_f32(
      /*neg_a=*/false, a, /*neg_b=*/false, b,
      /*c_mod=*/(short)0, c, /*reuse_a=*/false, /*reuse_b=*/false);
  float v = c[0];
  v += __shfl_xor(v, 1, 32);
  v += __shfl_xor(v, 2, 32);
  v += __shfl_xor(v, 4, 32);
  v += __shfl_xor(v, 8, 32);
  return v;  // full-wave total in every lane
}

__global__ __launch_bounds__(NTHREADS)
void total_registration_loss_kernel(const float* __restrict__ fixed_lm,
                                    const float* __restrict__ moving_lm,
                                    const float* __restrict__ df,
                                    const float* __restrict__ moving_sp,
                                    float* __restrict__ out) {
  __shared__ float wave_sums[NTHREADS / 32];

  const int tid  = threadIdx.x;
  const int lane = tid & 31;
  const int wave = tid >> 5;
  const int DHW  = DIM_D * DIM_H * DIM_W;

  const float spx = moving_sp[0];
  const float spy = moving_sp[1];
  const float spz = moving_sp[2];

  float acc = 0.0f;

  // 2048 landmarks / 1024 threads = uniform 2 iterations (EXEC stays full).
  #pragma unroll
  for (int i = tid; i < N_LM; i += NTHREADS) {
    const float mx = moving_lm[3 * i + 0];
    const float my = moving_lm[3 * i + 1];
    const float mz = moving_lm[3 * i + 2];
    const float fx = fixed_lm[3 * i + 0];
    const float fy = fixed_lm[3 * i + 1];
    const float fz = fixed_lm[3 * i + 2];

    const int fxi = (int)floorf(mx);
    const int fyi = (int)floorf(my);
    const int fzi = (int)floorf(mz);
    const int cxi = min((int)ceilf(mx), DIM_D - 1);
    const int cyi = min((int)ceilf(my), DIM_H - 1);
    const int czi = min((int)ceilf(mz), DIM_W - 1);

    const int idf = (fxi * DIM_H + fyi) * DIM_W + fzi;
    const int idc = (cxi * DIM_H + cyi) * DIM_W + czi;

    const float dx = 0.5f * (df[idf]           + df[idc]);
    const float dy = 0.5f * (df[DHW + idf]     + df[DHW + idc]);
    const float dz = 0.5f * (df[2 * DHW + idf] + df[2 * DHW + idc]);

    const float ex = (dx + mx - fx) * spx;
    const float ey = (dy + my - fy) * spy;
    const float ez = (dz + mz - fz) * spz;

    acc = fmaf(ex, ex, acc);
    acc = fmaf(ey, ey, acc);
    acc = fmaf(ez, ez, acc);
  }

  // Stage 1: per-wave reduction (WMMA) -> one float per wave into LDS.
  const float wsum = wave_reduce_wmma(acc);
  if (lane == 0) wave_sums[wave] = wsum;
  __syncthreads();

  // Stage 2: all waves redundantly reduce the 32 wave sums (uniform control
  // flow keeps EXEC fully populated for the WMMA).
  const float s = wave_sums[lane];
  const float total = wave_reduce_wmma(s);

  if (tid == 0) out[0] = sqrtf(total);
}

extern "C" void kernel_launch(void* const* d_in, const int* in_sizes, int n_in,
                              void* d_out, int out_size, void* d_ws, size_t ws_size,
                              hipStream_t stream) {
  // setup_inputs() order:
  //   0: fixed_landmarks   (2048*3) f32
  //   1: moving_landmarks  (2048*3) f32
  //   2: displacement_field(1,3,224,224,224) f32
  //   3: fixed_spacing     (3) f32  (unused by reference math)
  //   4: moving_spacing    (3) f32
  const float* fixed_lm  = (const float*)d_in[0];
  const float* moving_lm = (const float*)d_in[1];
  const float* df        = (const float*)d_in[2];
  const float* moving_sp = (const float*)d_in[4];
  float* out = (float*)d_out;

  total_registration_loss_kernel<<<1, NTHREADS, 0, stream>>>(
      fixed_lm, moving_lm, df, moving_sp, out);
}